// GCN_33449205301815
// MI455X (gfx1250) — compile-verified
//
#include <hip/hip_runtime.h>

#define NN 50000
#define NE 800000
#define HID 128

typedef __attribute__((ext_vector_type(2))) float v2f;
typedef __attribute__((ext_vector_type(8))) float v8f;

// ---------------- degree / normalization ----------------

__global__ void k_init_deg(float* __restrict__ deg) {
  int i = blockIdx.x * blockDim.x + threadIdx.x;
  if (i < NN) deg[i] = 1.0f;  // self-loop weight
}

__global__ void k_deg_acc(const int* __restrict__ col, const float* __restrict__ w,
                          float* __restrict__ deg) {
  int e = blockIdx.x * blockDim.x + threadIdx.x;
  if (e < NE) atomicAdd(&deg[col[e]], w[e]);
}

// dis = rsqrt(deg); also seed layer-1 scalar accumulator with the self-loop term
__global__ void k_dis_self(const float* __restrict__ deg, const float* __restrict__ x,
                           float* __restrict__ dis, float* __restrict__ s) {
  int i = blockIdx.x * blockDim.x + threadIdx.x;
  if (i >= NN) return;
  float d = deg[i];
  float r = (d > 0.0f) ? rsqrtf(d) : 0.0f;
  dis[i] = r;
  s[i] = r * r * x[i];  // self-loop: dis[i]*1*dis[i]*x[i]
}

// ---------------- layer 1 (rank-1 collapse) ----------------

// s[c] += dis[row]*w*dis[c] * x[row]   (4 bytes of scatter per edge)
__global__ void k_prop1(const int* __restrict__ row, const int* __restrict__ col,
                        const float* __restrict__ w, const float* __restrict__ dis,
                        const float* __restrict__ x, float* __restrict__ s) {
  int e = blockIdx.x * blockDim.x + threadIdx.x;
  if (e >= NE) return;
  int r = row[e], c = col[e];
  atomicAdd(&s[c], dis[r] * w[e] * dis[c] * x[r]);
}

// h[n,f] = relu(s[n]*W1[f] + b1[f])
__global__ void k_layer1(const float* __restrict__ s, const float* __restrict__ W1,
                         const float* __restrict__ b1, float* __restrict__ h) {
  int idx = blockIdx.x * blockDim.x + threadIdx.x;  // NN*HID threads
  int n = idx >> 7, f = idx & 127;
  float v = fmaf(s[n], W1[f], b1[f]);
  h[idx] = v > 0.0f ? v : 0.0f;
}

// ---------------- layer 2 GEMM: g = h @ W2 via fp32 WMMA ----------------
// One wave per 16x16 output tile; 8 waves/block cover the 8 N-tiles of one
// 16-row strip. 50000 = 3125*16 exactly -> no tails, EXEC all-ones.

__global__ void __launch_bounds__(256) k_gemm(const float* __restrict__ h,
                                              const float* __restrict__ W2,
                                              float* __restrict__ g) {
  const int lane  = threadIdx.x & 31;
  const int wave  = threadIdx.x >> 5;        // 0..7  -> N tile
  const int m0    = blockIdx.x * 16;         // 0..3124 -> M tile
  const int l16   = lane & 15;
  const int hi    = lane >> 4;               // 0: K={0,1}, 1: K={2,3}
  const int kb    = hi * 2;
  const int mrow  = m0 + l16;                // A row for this lane
  const int n     = wave * 16 + l16;         // B/C/D column for this lane

  const float* __restrict__ ha = h + (long)mrow * HID;
  v8f c = {};
#pragma unroll
  for (int k0 = 0; k0 < HID; k0 += 4) {
    v2f a, b;
    a.x = ha[k0 + kb];
    a.y = ha[k0 + kb + 1];
    b.x = W2[(k0 + kb) * HID + n];
    b.y = W2[(k0 + kb + 1) * HID + n];
    c = __builtin_amdgcn_wmma_f32_16x16x4_f32(false, a, false, b,
                                              (short)0, c, false, false);
  }
  const int mbase = m0 + hi * 8;
#pragma unroll
  for (int v = 0; v < 8; ++v)
    g[(long)(mbase + v) * HID + n] = c[v];
}

// ---------------- layer 2 propagation ----------------

// out[n,f] = b2[f] + dis[n]^2 * g[n,f]   (bias + self-loop, non-atomic init)
__global__ void k_out_init(const float* __restrict__ g, const float* __restrict__ dis,
                           const float* __restrict__ b2, float* __restrict__ out) {
  int idx = blockIdx.x * blockDim.x + threadIdx.x;  // NN*HID threads
  int n = idx >> 7, f = idx & 127;
  float r = dis[n];
  out[idx] = fmaf(r * r, g[idx], b2[f]);
}

// one wave32 per edge: each lane moves float4 (128 floats / 32 lanes)
__global__ void k_prop2(const int* __restrict__ row, const int* __restrict__ col,
                        const float* __restrict__ w, const float* __restrict__ dis,
                        const float* __restrict__ g, float* __restrict__ out) {
  int gid  = blockIdx.x * blockDim.x + threadIdx.x;
  int e    = gid >> 5;
  int lane = gid & 31;
  if (e >= NE) return;
  int r = row[e], c = col[e];
  float coeff = dis[r] * w[e] * dis[c];
  const float4* __restrict__ gr = (const float4*)(g + (long)r * HID);
  float4 v = gr[lane];
  float* oc = out + (long)c * HID + lane * 4;
  atomicAdd(oc + 0, coeff * v.x);
  atomicAdd(oc + 1, coeff * v.y);
  atomicAdd(oc + 2, coeff * v.z);
  atomicAdd(oc + 3, coeff * v.w);
}

// ---------------- launch ----------------

extern "C" void kernel_launch(void* const* d_in, const int* in_sizes, int n_in,
                              void* d_out, int out_size, void* d_ws, size_t ws_size,
                              hipStream_t stream) {
  const float* x    = (const float*)d_in[0];
  const int*   ei   = (const int*)d_in[1];
  const int*   row  = ei;            // edge_index[0] = source
  const int*   col  = ei + NE;       // edge_index[1] = target
  const float* w    = (const float*)d_in[2];
  const float* W1   = (const float*)d_in[3];
  const float* b1   = (const float*)d_in[4];
  const float* W2   = (const float*)d_in[5];
  const float* b2   = (const float*)d_in[6];
  float* out = (float*)d_out;

  // workspace layout (floats): dis[NN] | s[NN] | h[NN*HID] | g[NN*HID]
  float* dis = (float*)d_ws;
  float* s   = dis + NN;
  float* h   = s + NN;
  float* g   = h + (size_t)NN * HID;

  const int T = 256;
  const int gN  = (NN + T - 1) / T;
  const int gE  = (NE + T - 1) / T;
  const int gNF = (NN * HID) / T;          // 25000
  const int gEW = (NE * 32 + T - 1) / T;   // 100000 (wave per edge)

  k_init_deg<<<gN, T, 0, stream>>>(dis);                       // dis holds deg for now
  k_deg_acc <<<gE, T, 0, stream>>>(col, w, dis);
  k_dis_self<<<gN, T, 0, stream>>>(dis, x, dis, s);            // deg -> dis in place
  k_prop1   <<<gE, T, 0, stream>>>(row, col, w, dis, x, s);
  k_layer1  <<<gNF, T, 0, stream>>>(s, W1, b1, h);
  k_gemm    <<<NN / 16, T, 0, stream>>>(h, W2, g);             // 3125 blocks
  k_out_init<<<gNF, T, 0, stream>>>(g, dis, b2, out);
  k_prop2   <<<gEW, T, 0, stream>>>(row, col, w, dis, g, out);
}